// Net_77094662963210
// MI455X (gfx1250) — compile-verified
//
#include <hip/hip_runtime.h>

typedef __attribute__((ext_vector_type(2))) float v2f;
typedef __attribute__((ext_vector_type(8))) float v8f;

#define N_NODES 100000

// ---------------------------------------------------------------------------
// Degree / norm kernels
// ---------------------------------------------------------------------------
__global__ void init_deg_kernel(float* __restrict__ deg, int n) {
    int i = blockIdx.x * blockDim.x + threadIdx.x;
    if (i < n) deg[i] = 1.0f;   // self-loop contributes 1 to degree (gcn_norm)
}

__global__ void accum_deg_kernel(const long long* __restrict__ dst,
                                 float* __restrict__ deg, long long E) {
    long long e = (long long)blockIdx.x * blockDim.x + threadIdx.x;
    if (e < E) atomicAdd(&deg[(int)dst[e]], 1.0f);
}

__global__ void rsqrt_kernel(float* __restrict__ deg, int n) {
    int i = blockIdx.x * blockDim.x + threadIdx.x;
    if (i < n) {
        float d = deg[i];
        deg[i] = (d > 0.0f) ? rsqrtf(d) : 0.0f;   // deg >= 1 always
    }
}

// ---------------------------------------------------------------------------
// Dense GEMM h = X @ W using V_WMMA_F32_16X16X4_F32 (exact f32 path).
// One wave32 owns one 16x16 output tile; EXEC stays all-ones per WMMA rules.
// A(16x4 f32): lanes 0-15 hold rows M=0..15 with K={0,1}; lanes 16-31 K={2,3}.
// B(4x16 f32): VGPR0 lanes0-15 K=0, lanes16-31 K=2; VGPR1 K=1 / K=3.
// C(16x16 f32): VGPR r: lanes 0-15 -> M=r, lanes 16-31 -> M=r+8.
// ---------------------------------------------------------------------------
__global__ void gemm_wmma_f32(const float* __restrict__ X,
                              const float* __restrict__ W,
                              float* __restrict__ H,
                              int nrows, int K, int ncols) {
    const int lane   = threadIdx.x & 31;
    const int waveIB = threadIdx.x >> 5;
    const int tilesN = ncols >> 4;
    const int numTiles = (nrows >> 4) * tilesN;
    const int tile = blockIdx.x * (blockDim.x >> 5) + waveIB;
    if (tile >= numTiles) return;          // wave-uniform exit, EXEC stays full

    const int tm   = tile / tilesN;
    const int tn   = tile % tilesN;
    const int half = lane >> 4;            // 0 -> K pair {0,1}, 1 -> {2,3}
    const int l16  = lane & 15;
    const int arow = tm * 16 + l16;
    const int bcol = tn * 16 + l16;

    v8f acc = {};
    for (int k = 0; k < K; k += 4) {
        const int ka = k + half * 2;
        v2f a, b;
        a.x = X[(long long)arow * K + ka];
        a.y = X[(long long)arow * K + ka + 1];
        b.x = W[(long long)(ka + 0) * ncols + bcol];
        b.y = W[(long long)(ka + 1) * ncols + bcol];
        acc = __builtin_amdgcn_wmma_f32_16x16x4_f32(
            /*neg_a=*/false, a, /*neg_b=*/false, b,
            /*c_mod=*/(short)0, acc, /*reuse_a=*/false, /*reuse_b=*/false);
    }

    const int rbase = tm * 16 + half * 8;
#pragma unroll
    for (int r = 0; r < 8; ++r) {
        H[(long long)(rbase + r) * ncols + bcol] = acc[r];
    }
}

// ---------------------------------------------------------------------------
// Zero the aggregation buffer (scatter target).
// ---------------------------------------------------------------------------
__global__ void zero_kernel(float* __restrict__ p, long long n) {
    long long i = (long long)blockIdx.x * blockDim.x + threadIdx.x;
    if (i < n) p[i] = 0.0f;
}

// ---------------------------------------------------------------------------
// Edge scatter: out[dst] += h[src] * (disq[src]*disq[dst]).
// Thread handles 4 contiguous channels of one edge; consecutive lanes cover
// consecutive channel groups of the same edge -> coalesced float4 gather and
// contiguous global_atomic_add_f32 bursts (L2-resident: h <= 51MB << 192MB).
// ---------------------------------------------------------------------------
__global__ void edge_scatter_kernel(const float* __restrict__ H,
                                    const long long* __restrict__ src,
                                    const long long* __restrict__ dst,
                                    const float* __restrict__ disq,
                                    float* __restrict__ out,
                                    int dout, long long E) {
    const int cg = dout >> 2;                       // channel groups of 4
    long long idx = (long long)blockIdx.x * blockDim.x + threadIdx.x;
    long long total = E * (long long)cg;
    if (idx >= total) return;
    long long e = idx / cg;
    int c = (int)(idx - e * cg) << 2;

    const int s = (int)src[e];
    const int d = (int)dst[e];
    const float norm = disq[s] * disq[d];

    const float4 hv = *(const float4*)(H + (long long)s * dout + c);
    float* op = out + (long long)d * dout + c;
    atomicAdd(op + 0, hv.x * norm);
    atomicAdd(op + 1, hv.y * norm);
    atomicAdd(op + 2, hv.z * norm);
    atomicAdd(op + 3, hv.w * norm);
}

// ---------------------------------------------------------------------------
// Fused tail: y = relu(agg + self_norm * h + b), in-place on agg buffer.
// ---------------------------------------------------------------------------
__global__ void node_final_kernel(float* __restrict__ Y,      // agg in, y out
                                  const float* __restrict__ H,
                                  const float* __restrict__ disq,
                                  const float* __restrict__ bias,
                                  int dout, int n) {
    long long i = (long long)blockIdx.x * blockDim.x + threadIdx.x;
    long long total = (long long)n * dout;
    if (i >= total) return;
    int node = (int)(i / dout);
    int c    = (int)(i - (long long)node * dout);
    float di = disq[node];
    float v  = Y[i] + di * di * H[i] + bias[c];
    Y[i] = v > 0.0f ? v : 0.0f;
}

// ---------------------------------------------------------------------------
// Host-side orchestration
// ---------------------------------------------------------------------------
static inline unsigned cdiv_ll(long long a, long long b) {
    return (unsigned)((a + b - 1) / b);
}

extern "C" void kernel_launch(void* const* d_in, const int* in_sizes, int n_in,
                              void* d_out, int out_size, void* d_ws, size_t ws_size,
                              hipStream_t stream) {
    const float*     x  = (const float*)d_in[0];
    const long long* ei = (const long long*)d_in[1];
    const long long  E  = (long long)in_sizes[1] / 2;
    const long long* srcp = ei;
    const long long* dstp = ei + E;

    const float* Ws[4] = {(const float*)d_in[2], (const float*)d_in[4],
                          (const float*)d_in[6], (const float*)d_in[8]};
    const float* Bs[4] = {(const float*)d_in[3], (const float*)d_in[5],
                          (const float*)d_in[7], (const float*)d_in[9]};
    const int dins[4]  = {128, 64, 32, 64};
    const int douts[4] = {64, 32, 64, 128};

    float* out = (float*)d_out;
    const int N = N_NODES;

    // Workspace carve-up (256B aligned): disq[N], H[N*128], P[N*64], Q[N*32]
    char*  ws  = (char*)d_ws;
    size_t off = 0;
    float* disq = (float*)(ws + off); off += (((size_t)N * 4) + 255) & ~(size_t)255;
    float* Hbuf = (float*)(ws + off); off += (size_t)N * 128 * 4;
    float* Pbuf = (float*)(ws + off); off += (size_t)N * 64 * 4;
    float* Qbuf = (float*)(ws + off); off += (size_t)N * 32 * 4;
    (void)ws_size;

    // Aggregation/output buffer per layer (P, Q, P, d_out ping-pong)
    float* Ybufs[4] = {Pbuf, Qbuf, Pbuf, out};

    const int TB = 256;

    // 1) degree -> deg^{-1/2}
    init_deg_kernel<<<cdiv_ll(N, TB), TB, 0, stream>>>(disq, N);
    accum_deg_kernel<<<cdiv_ll(E, TB), TB, 0, stream>>>(dstp, disq, E);
    rsqrt_kernel<<<cdiv_ll(N, TB), TB, 0, stream>>>(disq, N);

    // 2) four GCN layers
    const float* xin = x;
    for (int l = 0; l < 4; ++l) {
        const int din  = dins[l];
        const int dout = douts[l];
        float* Y = Ybufs[l];

        // h = xin @ W  (WMMA f32, one wave per 16x16 tile, 8 waves/block)
        const int tiles = (N / 16) * (dout / 16);
        gemm_wmma_f32<<<cdiv_ll(tiles, TB / 32), TB, 0, stream>>>(
            xin, Ws[l], Hbuf, N, din, dout);

        // zero scatter target
        const long long ycount = (long long)N * dout;
        zero_kernel<<<cdiv_ll(ycount, TB), TB, 0, stream>>>(Y, ycount);

        // scatter-add messages over edges
        const long long work = E * (long long)(dout / 4);
        edge_scatter_kernel<<<cdiv_ll(work, TB), TB, 0, stream>>>(
            Hbuf, srcp, dstp, disq, Y, dout, E);

        // y = relu(agg + self_norm*h + b), in-place
        node_final_kernel<<<cdiv_ll(ycount, TB), TB, 0, stream>>>(
            Y, Hbuf, disq, Bs[l], dout, N);

        xin = Y;
    }
}